// NMS_without_score_thresholding_76656576299737
// MI455X (gfx1250) — compile-verified
//
#include <hip/hip_runtime.h>
#include <stdint.h>

#define N       8192
#define NW      128            // 64-bit mask words per row
#define TR      32             // rows per mask block
#define TC      2048           // cols per mask block
#define MASK_T  256            // threads in mask kernel
#define SORT_T  1024           // threads in sort kernel
#define PF      8              // scan prefetch depth (ring buffer rows)

// gfx1250 async global->LDS copy (per-lane 16B, tracked by ASYNCcnt)
__device__ __forceinline__ void async_b128_to_lds(uint32_t lds_off, const void* gptr) {
    asm volatile("global_load_async_to_lds_b128 %0, %1, off"
                 :: "v"(lds_off), "v"(gptr) : "memory");
}

// -------------------------------------------------------------------------
// Kernel 1: single-workgroup LDS bitonic sort of packed (conf, ~idx) keys.
// conf in [0,1) -> IEEE bits monotonic; tie-break ascending original index
// exactly matches jnp.argsort(-conf) stable semantics.
// -------------------------------------------------------------------------
__global__ __launch_bounds__(SORT_T)
void nms_sort_kernel(const float4* __restrict__ xyxy,
                     const float*  __restrict__ conf,
                     float4* __restrict__ sbox,
                     float*  __restrict__ sconf)
{
    __shared__ unsigned long long key[N];   // 64 KB LDS
    const int tid = threadIdx.x;

    for (int i = tid; i < N; i += SORT_T) {
        unsigned int cb = __float_as_uint(conf[i]);
        key[i] = ((unsigned long long)cb << 32) |
                 (unsigned long long)(0xFFFFFFFFu - (unsigned)i);
    }
    __syncthreads();

    for (int k = 2; k <= N; k <<= 1) {
        for (int j = k >> 1; j > 0; j >>= 1) {
            for (int i = tid; i < N; i += SORT_T) {
                int ixj = i ^ j;
                if (ixj > i) {
                    unsigned long long a = key[i];
                    unsigned long long b = key[ixj];
                    bool up = ((i & k) == 0);          // descending overall
                    if (up ? (a < b) : (a > b)) { key[i] = b; key[ixj] = a; }
                }
            }
            __syncthreads();
        }
    }

    for (int i = tid; i < N; i += SORT_T) {
        unsigned long long kk = key[i];
        unsigned idx = 0xFFFFFFFFu - (unsigned)(kk & 0xFFFFFFFFu);
        sbox[i]  = xyxy[idx];
        sconf[i] = __uint_as_float((unsigned)(kk >> 32));
    }
}

// -------------------------------------------------------------------------
// Kernel 2: suppression bitmask. Block = 32 rows x 2048 cols tile.
// Column boxes staged into LDS with async global->LDS copies, reused by
// 32 rows. Each thread-task produces one 64-bit word: bit b set iff
// (col > row && IoU > 0.5), exact division like the reference.
// -------------------------------------------------------------------------
__global__ __launch_bounds__(MASK_T)
void nms_mask_kernel(const float4* __restrict__ sbox,
                     unsigned long long* __restrict__ mask)
{
    __shared__ float4 cb[TC];   // 32 KB LDS column-tile
    const int tid = threadIdx.x;
    const int c0  = blockIdx.x * TC;
    const int r0  = blockIdx.y * TR;

    // async stage: 2048 boxes, 8 x B128 per thread (per-lane global->LDS DMA)
    for (int t = tid; t < TC; t += MASK_T) {
        async_b128_to_lds((uint32_t)(uintptr_t)(&cb[t]),
                          (const void*)(sbox + (c0 + t)));
    }
    asm volatile("s_wait_asynccnt 0x0" ::: "memory");
    __syncthreads();

    // 32 rows x 32 words = 1024 word-tasks, 4 per thread
    for (int t = tid; t < TR * (TC / 64); t += MASK_T) {
        const int rr  = t >> 5;       // row in tile
        const int wt  = t & 31;       // word in tile
        const int row = r0 + rr;

        float4 rb = sbox[row];
        float rArea = (rb.z - rb.x) * (rb.w - rb.y);

        unsigned long long word = 0ull;
        const int cbase = wt * 64;
        const int gc    = c0 + cbase;
        #pragma unroll 8
        for (int b = 0; b < 64; ++b) {
            float4 c  = cb[cbase + b];
            float lx  = fmaxf(rb.x, c.x), ly = fmaxf(rb.y, c.y);
            float hx  = fminf(rb.z, c.z), hy = fminf(rb.w, c.w);
            float w   = fmaxf(hx - lx, 0.0f);
            float h   = fmaxf(hy - ly, 0.0f);
            float inter = w * h;
            float cArea = (c.z - c.x) * (c.w - c.y);
            float uni   = rArea + cArea - inter;
            bool  sup   = ((inter / uni) > 0.5f) && ((gc + b) > row);
            if (sup) word |= (1ull << b);
        }
        mask[(size_t)row * NW + (size_t)(gc >> 6)] = word;
    }
}

// -------------------------------------------------------------------------
// Kernel 3: sequential greedy scan. One wave32, removed-bitmask (128 u64)
// in LDS, lockstep uniform branching -> no barriers. Mask rows are
// software-pipelined into an 8-deep LDS ring via async global->LDS copies:
// 2 x b128 per row (32 lanes x 16B = 512B each), consumption gated by
// s_wait_asynccnt 14 (= 2*(PF-1); async loads complete in order).
// -------------------------------------------------------------------------
__global__ __launch_bounds__(32)
void nms_scan_kernel(const unsigned long long* __restrict__ mask,
                     unsigned long long* __restrict__ removedOut)
{
    __shared__ unsigned long long rem[NW];        // 1 KB
    __shared__ unsigned long long rows[PF][NW];   // 8 KB ring buffer
    const int lane = threadIdx.x;                 // 0..31, single wave
    #pragma unroll
    for (int s = 0; s < NW / 32; ++s) rem[lane + 32 * s] = 0ull;

    // prologue: prefetch rows 0..PF-1
    for (int d = 0; d < PF; ++d) {
        #pragma unroll
        for (int s = 0; s < 2; ++s) {
            const int w = lane * 2 + 64 * s;
            async_b128_to_lds((uint32_t)(uintptr_t)(&rows[d][w]),
                              (const void*)(mask + (size_t)d * NW + w));
        }
    }

    for (int i = 0; i < N; ++i) {
        const int slot = i & (PF - 1);
        // retire the oldest row (row i): outstanding <= 2*(PF-1)
        asm volatile("s_wait_asynccnt 0xe" ::: "memory");

        const int w = i >> 6, bit = i & 63;
        unsigned long long rw = rem[w];                  // LDS broadcast (uniform)
        if (!((rw >> bit) & 1ull)) {                     // row i kept -> OR its row
            #pragma unroll
            for (int s = 0; s < NW / 32; ++s)
                rem[lane + 32 * s] |= rows[slot][lane + 32 * s];
        }

        // prefetch row i+PF into the slot just consumed (clamped tail is inert)
        int pf = i + PF; if (pf > N - 1) pf = N - 1;
        #pragma unroll
        for (int s = 0; s < 2; ++s) {
            const int ww = lane * 2 + 64 * s;
            async_b128_to_lds((uint32_t)(uintptr_t)(&rows[slot][ww]),
                              (const void*)(mask + (size_t)pf * NW + ww));
        }
    }

    #pragma unroll
    for (int s = 0; s < NW / 32; ++s)
        removedOut[lane + 32 * s] = rem[lane + 32 * s];
    // s_endpgm implicit wait-idle drains tail prefetches
}

// -------------------------------------------------------------------------
// Kernel 4: emit [N,5] = (box, score) * keep
// -------------------------------------------------------------------------
__global__ __launch_bounds__(256)
void nms_finalize_kernel(const float4* __restrict__ sbox,
                         const float*  __restrict__ sconf,
                         const unsigned long long* __restrict__ removed,
                         float* __restrict__ out)
{
    int i = blockIdx.x * blockDim.x + threadIdx.x;
    if (i >= N) return;
    bool kp = !((removed[i >> 6] >> (i & 63)) & 1ull);
    float m = kp ? 1.0f : 0.0f;
    float4 b = sbox[i];
    out[(size_t)i * 5 + 0] = b.x * m;
    out[(size_t)i * 5 + 1] = b.y * m;
    out[(size_t)i * 5 + 2] = b.z * m;
    out[(size_t)i * 5 + 3] = b.w * m;
    out[(size_t)i * 5 + 4] = sconf[i] * m;
}

extern "C" void kernel_launch(void* const* d_in, const int* in_sizes, int n_in,
                              void* d_out, int out_size, void* d_ws, size_t ws_size,
                              hipStream_t stream)
{
    (void)in_sizes; (void)n_in; (void)out_size; (void)ws_size;

    const float4* xyxy = (const float4*)d_in[0];   // [N,4] f32
    const float*  conf = (const float*)d_in[1];    // [N]   f32
    float* out = (float*)d_out;                    // [N,5] f32

    char* ws = (char*)d_ws;
    float4* sbox  = (float4*)(ws);                                   // 128 KB
    float*  sconf = (float*)(ws + 128 * 1024);                       // 32 KB
    unsigned long long* mask    = (unsigned long long*)(ws + 160 * 1024);          // 8 MB
    unsigned long long* removed = (unsigned long long*)(ws + 160 * 1024 + (size_t)N * NW * 8);

    nms_sort_kernel<<<1, SORT_T, 0, stream>>>(xyxy, conf, sbox, sconf);
    nms_mask_kernel<<<dim3(N / TC, N / TR), MASK_T, 0, stream>>>(sbox, mask);
    nms_scan_kernel<<<1, 32, 0, stream>>>(mask, removed);
    nms_finalize_kernel<<<(N + 255) / 256, 256, 0, stream>>>(sbox, sconf, removed, out);
}